// DifferentiableHodgeProxy_53291954208929
// MI455X (gfx1250) — compile-verified
//
#include <hip/hip_runtime.h>
#include <hip/hip_bf16.h>
#include <math.h>

typedef __attribute__((ext_vector_type(2))) float    v2f;
typedef __attribute__((ext_vector_type(8))) float    v8f;
typedef __attribute__((ext_vector_type(4))) unsigned v4u;
typedef __attribute__((ext_vector_type(8))) unsigned v8u;

#define BATCH 256
#define NPTS  4096
#define LIFT  16
#define KP    16      // MAXP
#define NS    3
#define NE    120
#define NT    560
#define HID   256
#define FEATP 32      // 28 features padded to 32
#define NJ    4
#define TAU   1e-4f
#define CHK   80      // k-chunk for term_up GEMM (560 = 7*80)

__device__ __forceinline__ v8f wmma4(v2f a, v2f b, v8f c) {
  // V_WMMA_F32_16X16X4_F32 : D = A(16x4) * B(4x16) + C(16x16), all f32
  return __builtin_amdgcn_wmma_f32_16x16x4_f32(
      /*neg_a=*/false, a, /*neg_b=*/false, b,
      /*c_mod=*/(short)0, c, /*reuse_a=*/false, /*reuse_b=*/false);
}

// ---------------------------------------------------------------------------
// TDM: DMA a 2D tile (tile_dim0=CHK cols, tile_dim1=NE rows, f32) of B2
// from global memory into LDS. D# per CDNA5 ISA ch.8 (group0 128b, group1
// 256b, groups 2/3 zero for a 2D tensor). Issued by one wave; completion via
// TENSORcnt.
// ---------------------------------------------------------------------------
__device__ __forceinline__ void tdm_load_tile_f32(const float* gsrc,
                                                  unsigned lds_byte_off) {
  unsigned long long ga = (unsigned long long)(uintptr_t)gsrc;
  v4u g0;
  g0[0] = 1u;                                        // count=1, user descriptor
  g0[1] = lds_byte_off;                              // lds_addr
  g0[2] = (unsigned)(ga & 0xFFFFFFFFu);              // global_addr[31:0]
  g0[3] = (unsigned)((ga >> 32) & 0x01FFFFFFu)       // global_addr[56:32]
        | (2u << 30);                                // type = 2 ("image")
  v8u g1;
  g1[0] = (2u << 16);                                // data_size = 4B; mask=0
  g1[1] = ((unsigned)NT & 0xFFFFu) << 16;            // tensor_dim0[15:0]
  g1[2] = (((unsigned)NT >> 16) & 0xFFFFu)           // tensor_dim0[31:16]
        | (((unsigned)NE & 0xFFFFu) << 16);          // tensor_dim1[15:0]
  g1[3] = (((unsigned)NE >> 16) & 0xFFFFu)           // tensor_dim1[31:16]
        | ((unsigned)CHK << 16);                     // tile_dim0
  g1[4] = (unsigned)NE;                              // tile_dim1 (tile_dim2=0)
  g1[5] = (unsigned)NT;                              // tensor_dim0_stride[31:0]
  g1[6] = 0u;                                        // stride hi / dim1_stride lo
  g1[7] = 0u;
  v4u gz = {0u, 0u, 0u, 0u};                         // groups 2/3: unused (2D)
  asm volatile("tensor_load_to_lds %0, %1, %2, %3"
               :
               : "s"(g0), "s"(g1), "s"(gz), "s"(gz)
               : "memory");
}

// ---------------------------------------------------------------------------
// Kernel 1: per-batch top-16, gather, distances, stats, W1/W2 weights
// ---------------------------------------------------------------------------
__global__ void topk_geom_kernel(const float* __restrict__ cloud_g,   // B,N,LIFT
                                 const float* __restrict__ ystar,     // B,N
                                 const float* __restrict__ lsc,       // S
                                 const int* __restrict__ ei, const int* __restrict__ ej,
                                 const int* __restrict__ tij, const int* __restrict__ tjk,
                                 const int* __restrict__ tik,
                                 float* __restrict__ ws_act,
                                 float* __restrict__ ws_W1,
                                 float* __restrict__ ws_W2,
                                 float* __restrict__ ws_raw)
{
  const int b   = blockIdx.x;
  const int tid = threadIdx.x;     // 256 threads = 8 waves

  __shared__ float ys[NPTS];       // staged weights: global read once
  __shared__ float rv[256];
  __shared__ int   ri[256];
  __shared__ int   chosen[KP];
  __shared__ float chosenv[KP];
  __shared__ float cl[KP][LIFT];
  __shared__ float Dm[KP][KP];
  __shared__ float msk[KP];
  __shared__ float W1s[NS][NE];
  __shared__ float red[256];
  __shared__ float scl[NS];

  const float* y = ystar + (size_t)b * NPTS;
  for (int t = tid; t < NPTS; t += 256) ys[t] = y[t];
  __syncthreads();

  // ---- iterative top-16: argmax + mark-selected with -inf ----
  for (int sel = 0; sel < KP; ++sel) {
    float bv = -INFINITY; int bi = 0;
    for (int t = tid; t < NPTS; t += 256) {
      float v = ys[t];
      if (v > bv) { bv = v; bi = t; }
    }
    rv[tid] = bv; ri[tid] = bi;
    __syncthreads();
    for (int off = 128; off > 0; off >>= 1) {
      if (tid < off) {
        if (rv[tid+off] > rv[tid] ||
            (rv[tid+off] == rv[tid] && ri[tid+off] < ri[tid])) {
          rv[tid] = rv[tid+off]; ri[tid] = ri[tid+off];
        }
      }
      __syncthreads();
    }
    if (tid == 0) {
      chosen[sel] = ri[0]; chosenv[sel] = rv[0];
      ys[ri[0]] = -INFINITY;                 // exclude from later passes
    }
    __syncthreads();
  }

  // ---- mask, act, gather ----
  if (tid < KP) {
    float a = chosenv[tid];
    msk[tid] = (a > 0.001f) ? 1.0f : 0.0f;
    ws_act[b*KP + tid] = a;
  }
  if (tid < NS) scl[tid] = expf(lsc[tid]);
  __syncthreads();
  if (tid < KP * LIFT) {
    int k = tid / LIFT, d = tid % LIFT;
    cl[k][d] = cloud_g[((size_t)b * NPTS + (size_t)chosen[k]) * LIFT + d];
  }
  __syncthreads();

  // ---- masked pairwise distances ----
  if (tid < KP*KP) {
    int k = tid / KP, l = tid % KP;
    float d2 = 0.f;
    for (int d = 0; d < LIFT; ++d) { float df = cl[k][d]-cl[l][d]; d2 += df*df; }
    float dd = (d2 > 0.f) ? sqrtf(d2) : 0.f;   // sqrt(where(d2>0,d2,1))*(d2>0)
    Dm[k][l] = dd * msk[k] * msk[l];
  }
  __syncthreads();

  // ---- upper-triangle stats ----
  float tri, dval;
  {
    int k2 = tid / KP, l2 = tid % KP;
    tri  = (k2 < l2) ? msk[k2]*msk[l2] : 0.f;
    dval = (k2 < l2) ? Dm[k2][l2]      : 0.f;
  }
  red[tid] = tri; __syncthreads();
  for (int off=128; off>0; off>>=1) { if (tid<off) red[tid]+=red[tid+off]; __syncthreads(); }
  float tsum = fmaxf(red[0], 1.0f); __syncthreads();

  red[tid] = dval*tri; __syncthreads();
  for (int off=128; off>0; off>>=1) { if (tid<off) red[tid]+=red[tid+off]; __syncthreads(); }
  float dsum = red[0]; __syncthreads();

  red[tid] = dval*tri; __syncthreads();
  for (int off=128; off>0; off>>=1) { if (tid<off) red[tid]=fmaxf(red[tid],red[tid+off]); __syncthreads(); }
  float dmax = red[0]; __syncthreads();

  float mean = dsum / tsum;
  red[tid] = (dval-mean)*(dval-mean)*tri; __syncthreads();
  for (int off=128; off>0; off>>=1) { if (tid<off) red[tid]+=red[tid+off]; __syncthreads(); }
  float vsum = red[0]; __syncthreads();

  if (tid == 0) {
    float* raw = ws_raw + b*FEATP;
    raw[24] = mean;
    raw[25] = dmax;
    raw[26] = vsum / tsum;
    raw[27] = mean / (dmax + 1e-6f);
    raw[28] = 0.f; raw[29] = 0.f; raw[30] = 0.f; raw[31] = 0.f;
  }

  // ---- W1 = exp(-D^2/denom) * mask2d at edges ----
  for (int x = tid; x < NS*NE; x += 256) {
    int s = x / NE, e = x % NE;
    int i = ei[e], j = ej[e];
    float d = Dm[i][j];
    float denom = 2.f*scl[s]*scl[s] + 1e-8f;
    float w = expf(-(d*d)/denom) * msk[i]*msk[j];
    W1s[s][e] = w;
    ws_W1[((size_t)b*NS + s)*NE + e] = w;
  }
  __syncthreads();

  // ---- W2 = W1[t_ij]*W1[t_jk]*W1[t_ik] ----
  for (int x = tid; x < NS*NT; x += 256) {
    int s = x / NT, t = x % NT;
    ws_W2[((size_t)b*NS + s)*NT + t] = W1s[s][tij[t]] * W1s[s][tjk[t]] * W1s[s][tik[t]];
  }
}

// ---------------------------------------------------------------------------
// Kernel 2: L0 (16x16 Laplacian) build + parallel Jacobi eigenvalues
// one block of 64 threads per (b,s)
// ---------------------------------------------------------------------------
__global__ void eig0_kernel(const float* __restrict__ ws_W1,
                            const int* __restrict__ ei, const int* __restrict__ ej,
                            float* __restrict__ ws_raw)
{
  const int bs = blockIdx.x, b = bs / NS, s = bs % NS;
  const int tid = threadIdx.x;   // 64
  __shared__ float A[KP][KP+1];
  __shared__ float jc[8], js[8];
  __shared__ int   jp[8], jq[8];
  const float* W1 = ws_W1 + (size_t)bs * NE;

  for (int x = tid; x < KP*KP; x += 64) {
    int k = x / KP, l = x % KP;
    A[k][l] = 0.f;
  }
  __syncthreads();
  // off-diagonals: unique edge per unordered pair (complete graph on 16)
  for (int e = tid; e < NE; e += 64) {
    int i = ei[e], j = ej[e]; float w = W1[e];
    A[i][j] = -w; A[j][i] = -w;
  }
  __syncthreads();
  if (tid < KP) {
    float dsum = TAU;
    for (int e = 0; e < NE; ++e)
      if (ei[e] == tid || ej[e] == tid) dsum += W1[e];
    A[tid][tid] = dsum;
  }
  __syncthreads();

  // cyclic parallel Jacobi, n=16: 15 rounds x 8 disjoint pairs
  for (int sweep = 0; sweep < 10; ++sweep) {
    for (int r = 0; r < 15; ++r) {
      if (tid < 8) {
        int j  = tid;
        int a1 = (j == 0) ? 0 : (1 + ((j - 1 + r) % 15));
        int a2 = 1 + ((14 - j + r) % 15);
        int p  = a1 < a2 ? a1 : a2;
        int q  = a1 < a2 ? a2 : a1;
        float app = A[p][p], aqq = A[q][q], apq = A[p][q];
        float c = 1.f, sn = 0.f;
        if (fabsf(apq) > 1e-12f) {
          float th = (aqq - app) / (2.f * apq);
          float t  = ((th >= 0.f) ? 1.f : -1.f) / (fabsf(th) + sqrtf(1.f + th*th));
          c  = 1.f / sqrtf(1.f + t*t);
          sn = t * c;
        }
        jc[j] = c; js[j] = sn; jp[j] = p; jq[j] = q;
      }
      __syncthreads();
      for (int x = tid; x < 8*KP; x += 64) {
        int j = x / KP, t = x % KP;
        int p = jp[j], q = jq[j]; float c = jc[j], sn = js[j];
        float xp = A[p][t], xq = A[q][t];
        A[p][t] = c*xp - sn*xq; A[q][t] = sn*xp + c*xq;
      }
      __syncthreads();
      for (int x = tid; x < 8*KP; x += 64) {
        int j = x / KP, t = x % KP;
        int p = jp[j], q = jq[j]; float c = jc[j], sn = js[j];
        float xp = A[t][p], xq = A[t][q];
        A[t][p] = c*xp - sn*xq; A[t][q] = sn*xp + c*xq;
      }
      __syncthreads();
    }
  }

  if (tid == 0) {
    float ev[KP];
    for (int k = 0; k < KP; ++k) ev[k] = A[k][k];
    float* out = ws_raw + b*FEATP + s*8;       // eig0 slots
    for (int a = 0; a < NJ; ++a) {
      int m = a;
      for (int k = a+1; k < KP; ++k) if (ev[k] < ev[m]) m = k;
      float t = ev[a]; ev[a] = ev[m]; ev[m] = t;
      out[a] = ev[a];
    }
  }
}

// ---------------------------------------------------------------------------
// Kernel 3: L1 = term_down + B2 diag(W2) B2^T + tau I  (120x120, in LDS).
// B2 chunks DMA'd global->LDS by the Tensor Data Mover, scaled in place by
// sqrt(W2), term_up via f32 WMMA (M M^T), then parallel Jacobi eigensolve.
// one block of 256 threads (8 waves) per (b,s)
// ---------------------------------------------------------------------------
__global__ void l1_kernel(const float* __restrict__ ws_W2,
                          const float* __restrict__ ws_act,
                          const float* __restrict__ B2g,        // 120x560
                          const int* __restrict__ ei, const int* __restrict__ ej,
                          float* __restrict__ ws_raw)
{
  const int bs = blockIdx.x, b = bs / NS, s = bs % NS;
  const int tid  = threadIdx.x;          // 256
  const int wv   = tid >> 5;             // wave id 0..7
  const int lane = tid & 31;
  const int half = (lane >> 4) & 1;
  const int n    = lane & 15;            // tile row/col within 16

  __shared__ float L[NE * 124];          // 120x120 padded stride 124
  __shared__ float Mc[128 * CHK];        // staged M chunk (8 zero pad rows)
  __shared__ float acts[KP];
  __shared__ float jc[60], js[60];
  __shared__ int   jp[60], jq[60];

  const float* W2g = ws_W2 + (size_t)bs * NT;

  if (tid < KP) acts[tid] = ws_act[b*KP + tid];
  // zero pad rows 120..127 once (TDM only writes rows 0..119)
  for (int x = tid; x < 8*CHK; x += 256) Mc[NE*CHK + x] = 0.f;
  __syncthreads();

  // ---- term_down + tau I, via B1 column sparsity (2 nonzeros/column) ----
  for (int x = tid; x < NE*NE; x += 256) {
    int e = x / NE, f = x % NE;
    int ie = ei[e], je = ej[e], iff = ei[f], jf = ej[f];
    float v = 0.f;
    if (ie == iff) v += acts[ie];
    if (ie == jf)  v -= acts[ie];
    if (je == iff) v -= acts[je];
    if (je == jf)  v += acts[je];
    if (e == f)    v += TAU;
    L[e*124 + f] = v;
  }

  const unsigned mc_off = (unsigned)(uintptr_t)(void*)&Mc[0];

  // ---- term_up = M M^T with M = B2 .* sqrt(W2), f32 WMMA 16x16x4 ----
  v8f acc[8] = {};
  for (int c = 0; c < NT / CHK; ++c) {
    __syncthreads();                     // all waves done reading prev chunk
    if (wv == 0) {
      tdm_load_tile_f32(B2g + c*CHK, mc_off);   // raw B2 tile -> LDS via TDM
      __builtin_amdgcn_s_wait_tensorcnt(0);
    }
    __syncthreads();                     // tile visible to all waves
    // in-place scale by sqrt(W2) (rows < 120)
    for (int x = tid; x < NE*CHK; x += 256) {
      int kc = x % CHK;
      Mc[x] *= sqrtf(W2g[c*CHK + kc]);
    }
    __syncthreads();
    const int arow = wv*16 + n;
    for (int ct = 0; ct < 8; ++ct) {
      const int brow = ct*16 + n;
      for (int kk = 0; kk < CHK; kk += 4) {
        v2f av, bv;
        av.x = Mc[arow*CHK + kk + 2*half];
        av.y = Mc[arow*CHK + kk + 2*half + 1];
        bv.x = Mc[brow*CHK + kk + 2*half];
        bv.y = Mc[brow*CHK + kk + 2*half + 1];
        acc[ct] = wmma4(av, bv, acc[ct]);
      }
    }
  }
  __syncthreads();
  // scatter accumulators into L (unique owner per element)
  for (int ct = 0; ct < 8; ++ct) {
    for (int r = 0; r < 8; ++r) {
      int row = wv*16 + r + 8*half;
      int col = ct*16 + n;
      if (row < NE && col < NE) L[row*124 + col] += acc[ct][r];
    }
  }
  __syncthreads();

  // ---- parallel cyclic Jacobi, n=120: 119 rounds x 60 disjoint pairs ----
  for (int sweep = 0; sweep < 6; ++sweep) {
    for (int r = 0; r < 119; ++r) {
      if (tid < 60) {
        int j  = tid;
        int a1 = (j == 0) ? 0 : (1 + ((j - 1 + r) % 119));
        int a2 = 1 + ((118 - j + r) % 119);
        int p  = a1 < a2 ? a1 : a2;
        int q  = a1 < a2 ? a2 : a1;
        float app = L[p*124+p], aqq = L[q*124+q], apq = L[p*124+q];
        float c = 1.f, sn = 0.f;
        if (fabsf(apq) > 1e-12f) {
          float th = (aqq - app) / (2.f * apq);
          float t  = ((th >= 0.f) ? 1.f : -1.f) / (fabsf(th) + sqrtf(1.f + th*th));
          c  = 1.f / sqrtf(1.f + t*t);
          sn = t * c;
        }
        jc[j] = c; js[j] = sn; jp[j] = p; jq[j] = q;
      }
      __syncthreads();
      for (int x = tid; x < 60*NE; x += 256) {
        int j = x / NE, t = x % NE;
        int p = jp[j], q = jq[j]; float c = jc[j], sn = js[j];
        float xp = L[p*124 + t], xq = L[q*124 + t];
        L[p*124 + t] = c*xp - sn*xq; L[q*124 + t] = sn*xp + c*xq;
      }
      __syncthreads();
      for (int x = tid; x < 60*NE; x += 256) {
        int j = x / NE, t = x % NE;
        int p = jp[j], q = jq[j]; float c = jc[j], sn = js[j];
        float xp = L[t*124 + p], xq = L[t*124 + q];
        L[t*124 + p] = c*xp - sn*xq; L[t*124 + q] = sn*xp + c*xq;
      }
      __syncthreads();
    }
  }

  // ---- 4 smallest eigenvalues (ascending) ----
  if (tid == 0) {
    int usedi[NJ];
    float* out = ws_raw + b*FEATP + s*8 + 4;   // eig1 slots
    for (int a = 0; a < NJ; ++a) {
      float mv = INFINITY; int mi = 0;
      for (int k = 0; k < NE; ++k) {
        bool u = false;
        for (int c = 0; c < a; ++c) u = u || (usedi[c] == k);
        float v = L[k*124 + k];
        if (!u && v < mv) { mv = v; mi = k; }
      }
      usedi[a] = mi; out[a] = mv;
    }
  }
}

// ---------------------------------------------------------------------------
// Kernel 4: h = gelu(raw @ w1 + b1), f32 WMMA, one wave per 16x16 tile
// ---------------------------------------------------------------------------
__global__ void mlp1_kernel(const float* __restrict__ raw,  // 256 x FEATP
                            const float* __restrict__ w1,   // 28 x 256
                            const float* __restrict__ b1,
                            float* __restrict__ h)
{
  const int lane = threadIdx.x;
  const int half = lane >> 4, n = lane & 15;
  const int mBase = blockIdx.y * 16, nBase = blockIdx.x * 16;
  v8f acc = {};
  for (int k = 0; k < 28; k += 4) {
    v2f a, bb;
    a.x  = raw[(mBase + n)*FEATP + k + 2*half];
    a.y  = raw[(mBase + n)*FEATP + k + 2*half + 1];
    bb.x = w1[(k + 2*half)*HID + nBase + n];
    bb.y = w1[(k + 2*half + 1)*HID + nBase + n];
    acc = wmma4(a, bb, acc);
  }
  float bias = b1[nBase + n];
  for (int r = 0; r < 8; ++r) {
    int row = mBase + r + 8*half;
    float x = acc[r] + bias;
    float g = 0.5f * x * (1.0f + erff(x * 0.70710678f));   // exact gelu
    h[row*HID + nBase + n] = g;
  }
}

// ---------------------------------------------------------------------------
// Kernel 5: out = h @ w2 + b2, f32 WMMA
// ---------------------------------------------------------------------------
__global__ void mlp2_kernel(const float* __restrict__ h,
                            const float* __restrict__ w2,
                            const float* __restrict__ b2,
                            float* __restrict__ out)
{
  const int lane = threadIdx.x;
  const int half = lane >> 4, n = lane & 15;
  const int mBase = blockIdx.y * 16, nBase = blockIdx.x * 16;
  v8f acc = {};
  for (int k = 0; k < HID; k += 4) {
    v2f a, bb;
    a.x  = h[(mBase + n)*HID + k + 2*half];
    a.y  = h[(mBase + n)*HID + k + 2*half + 1];
    bb.x = w2[(k + 2*half)*HID + nBase + n];
    bb.y = w2[(k + 2*half + 1)*HID + nBase + n];
    acc = wmma4(a, bb, acc);
  }
  float bias = b2[nBase + n];
  for (int r = 0; r < 8; ++r)
    out[(mBase + r + 8*half)*HID + nBase + n] = acc[r] + bias;
}

// ---------------------------------------------------------------------------
extern "C" void kernel_launch(void* const* d_in, const int* in_sizes, int n_in,
                              void* d_out, int out_size, void* d_ws, size_t ws_size,
                              hipStream_t stream) {
  (void)in_sizes; (void)n_in; (void)out_size; (void)ws_size;
  const float* dense = (const float*)d_in[0];
  const float* ystar = (const float*)d_in[1];
  const float* lsc   = (const float*)d_in[2];
  const float* w1    = (const float*)d_in[3];
  const float* b1    = (const float*)d_in[4];
  const float* w2    = (const float*)d_in[5];
  const float* b2    = (const float*)d_in[6];
  /* d_in[7] = B1 : unused, incidence structure recovered from edge lists */
  const float* B2g   = (const float*)d_in[8];
  const int* ei  = (const int*)d_in[9];
  const int* ej  = (const int*)d_in[10];
  const int* tij = (const int*)d_in[11];
  const int* tjk = (const int*)d_in[12];
  const int* tik = (const int*)d_in[13];

  char* ws = (char*)d_ws;
  float* ws_act = (float*)(ws);                                   // 256*16*4      = 16384
  float* ws_W1  = (float*)(ws + 16384);                           // 256*3*120*4   = 368640
  float* ws_W2  = (float*)(ws + 16384 + 368640);                  // 256*3*560*4   = 1720320
  float* ws_raw = (float*)(ws + 16384 + 368640 + 1720320);        // 256*32*4      = 32768
  float* ws_h   = (float*)(ws + 16384 + 368640 + 1720320 + 32768);// 256*256*4     = 262144

  topk_geom_kernel<<<dim3(BATCH), dim3(256), 0, stream>>>(
      dense, ystar, lsc, ei, ej, tij, tjk, tik, ws_act, ws_W1, ws_W2, ws_raw);
  eig0_kernel<<<dim3(BATCH*NS), dim3(64), 0, stream>>>(ws_W1, ei, ej, ws_raw);
  l1_kernel<<<dim3(BATCH*NS), dim3(256), 0, stream>>>(ws_W2, ws_act, B2g, ei, ej, ws_raw);
  mlp1_kernel<<<dim3(16,16), dim3(32), 0, stream>>>(ws_raw, w1, b1, ws_h);
  mlp2_kernel<<<dim3(16,16), dim3(32), 0, stream>>>(ws_h, w2, b2, (float*)d_out);
}